// BertCRFClassifier_7344394076472
// MI455X (gfx1250) — compile-verified
//
#include <hip/hip_runtime.h>
#include <hip/hip_bf16.h>

// Problem constants (from reference): B=64, L=512, H=768, T=9
#define BATCH 64
#define SEQ   512
#define HID   768
#define NTAG  9
#define NPAD  16   // T padded to WMMA N=16

typedef __attribute__((ext_vector_type(2))) float v2f;
typedef __attribute__((ext_vector_type(8))) float v8f;

// ---------------------------------------------------------------------------
// Kernel 1: logits[M=B*L, N=9] = hidden[M, 768] @ W[768, 9] + b
// fp32 WMMA 16x16x4. One wave = one 16-row M tile; 8 waves / block.
// W is staged into LDS padded to 16 columns, interleaved as
//   ldsB[kpair][n][0..1]  (kpair = k/2) so each B-fragment is one ds_load_b64.
// ---------------------------------------------------------------------------
__global__ __launch_bounds__(256) void bertcrf_logits_wmma(
    const float* __restrict__ hidden,   // [B*L, 768]
    const float* __restrict__ W,        // [768, 9]
    const float* __restrict__ bias,     // [9]
    float* __restrict__ logits)         // [B*L, 9]
{
    __shared__ float ldsB[(HID / 2) * (NPAD * 2)];   // 384*32 floats = 48 KB

    const int tid = threadIdx.x;

    // Cooperative staged load of W, zero-padded to 16 columns.
    for (int idx = tid; idx < HID * NPAD; idx += 256) {
        const int k = idx >> 4;
        const int n = idx & 15;
        ldsB[(k >> 1) * 32 + n * 2 + (k & 1)] =
            (n < NTAG) ? W[k * NTAG + n] : 0.0f;
    }
    __syncthreads();

    const int wave = tid >> 5;
    const int lane = tid & 31;
    const int tile = blockIdx.x * 8 + wave;      // 0 .. 2047 (B*L/16 tiles)
    const int m    = lane & 15;                  // row within tile (A) / col (B,C)
    const int kh   = (lane >> 4) << 1;           // 0 or 2 (K sub-offset per ISA layout)

    const float* aptr = hidden + ((size_t)tile * 16 + m) * HID + kh;
    const int    bofs = (lane & 15) * 2;

    v8f acc = {0.f, 0.f, 0.f, 0.f, 0.f, 0.f, 0.f, 0.f};

#pragma unroll 8
    for (int k = 0; k < HID; k += 4) {
        // A fragment: lane m holds hidden[row=m][k+kh], [k+kh+1]  (even pair -> b64)
        v2f a = *(const v2f*)(aptr + k);
        // B fragment: lane n holds W[k+kh][n], W[k+kh+1][n]
        v2f bb = *(const v2f*)(ldsB + ((k + kh) >> 1) * 32 + bofs);
        acc = __builtin_amdgcn_wmma_f32_16x16x4_f32(
            /*neg_a=*/false, a, /*neg_b=*/false, bb,
            /*c_mod=*/(short)0, acc, /*reuse_a=*/false, /*reuse_b=*/false);
    }

    // C/D layout: VGPR r -> row = r + (lane<16 ? 0 : 8), col = lane & 15
    const int col = lane & 15;
    if (col < NTAG) {
        const float bv = bias[col];
        const size_t row0 = (size_t)tile * 16 + ((lane >> 4) * 8);
#pragma unroll
        for (int r = 0; r < 8; ++r) {
            logits[(row0 + r) * NTAG + col] = acc[r] + bv;
        }
    }
}

// ---------------------------------------------------------------------------
// Kernel 2: CRF numerator + forward (alpha) scan + per-batch log-likelihood.
// One wave (32 lanes) per batch. Lane j owns tag j; cross-lane score access
// via __shfl (wave32). Serial over t (true data dependence), parallel over
// batch x tag. Native v_exp/v_log transcendentals.
// ---------------------------------------------------------------------------
__global__ __launch_bounds__(32) void bertcrf_forward(
    const float* __restrict__ logits,      // [B, L, 9]
    const int*   __restrict__ attn_mask,   // [B, L]
    const int*   __restrict__ labels,      // [B, L]
    const float* __restrict__ start_trans, // [9]
    const float* __restrict__ end_trans,   // [9]
    const float* __restrict__ trans,       // [9, 9]
    float* __restrict__ llh)               // [B]
{
    const int b    = blockIdx.x;
    const int lane = threadIdx.x;
    const int jj   = (lane < NTAG) ? lane : 0;

    const float* lg = logits + (size_t)b * SEQ * NTAG;
    const int*   mk = attn_mask + b * SEQ;
    const int*   lb = labels + b * SEQ;

    // Cache transition column j: tcol[i] = trans[i -> j]
    float tcol[NTAG];
#pragma unroll
    for (int i = 0; i < NTAG; ++i) tcol[i] = trans[i * NTAG + jj];

    // ---- numerator: strided over lanes, shfl reduction ----
    float part = 0.0f;
    int   cnt  = 0;
    for (int t = lane; t < SEQ; t += 32) {
        const int m = mk[t];
        cnt += m;
        if (t >= 1 && m) {
            int lp = lb[t - 1]; if (lp == -100) lp = 0;
            int lc = lb[t];     if (lc == -100) lc = 0;
            part += trans[lp * NTAG + lc] + lg[t * NTAG + lc];
        }
    }
#pragma unroll
    for (int off = 16; off > 0; off >>= 1) {
        part += __shfl_xor(part, off, 32);
        cnt  += __shfl_xor(cnt,  off, 32);
    }
    int l0 = lb[0]; if (l0 == -100) l0 = 0;
    float num = start_trans[l0] + lg[l0] + part;
    int send = cnt - 1; if (send < 0) send = 0;
    int lastT = lb[send]; if (lastT == -100) lastT = 0;
    num += end_trans[lastT];

    // ---- forward algorithm (log-semiring scan over t) ----
    float score = start_trans[jj] + lg[jj];
    for (int t = 1; t < SEQ; ++t) {
        const float e = lg[t * NTAG + jj];
        const int   m = mk[t];
        float s[NTAG];
#pragma unroll
        for (int i = 0; i < NTAG; ++i)
            s[i] = __shfl(score, i, 32) + tcol[i];
        float mx = s[0];
#pragma unroll
        for (int i = 1; i < NTAG; ++i) mx = fmaxf(mx, s[i]);
        float sum = 0.0f;
#pragma unroll
        for (int i = 0; i < NTAG; ++i) sum += __expf(s[i] - mx);
        const float nxt = mx + __logf(sum) + e;
        score = m ? nxt : score;
    }

    // ---- denominator: logsumexp_j(score_j + end_trans[j]) ----
    float v = (lane < NTAG) ? (score + end_trans[lane]) : -3.0e38f;
    float mx = v;
#pragma unroll
    for (int off = 16; off > 0; off >>= 1)
        mx = fmaxf(mx, __shfl_xor(mx, off, 32));
    float ex = __expf(v - mx);
#pragma unroll
    for (int off = 16; off > 0; off >>= 1)
        ex += __shfl_xor(ex, off, 32);
    const float denom = mx + __logf(ex);

    if (lane == 0) llh[b] = num - denom;
}

// ---------------------------------------------------------------------------
// Kernel 3: loss = -mean(llh)  (64 values, one wave)
// ---------------------------------------------------------------------------
__global__ __launch_bounds__(32) void bertcrf_loss(
    const float* __restrict__ llh, float* __restrict__ out)
{
    const int lane = threadIdx.x;
    float v = llh[lane] + llh[lane + 32];
#pragma unroll
    for (int off = 16; off > 0; off >>= 1)
        v += __shfl_xor(v, off, 32);
    if (lane == 0) out[0] = -v / (float)BATCH;
}

// ---------------------------------------------------------------------------
extern "C" void kernel_launch(void* const* d_in, const int* in_sizes, int n_in,
                              void* d_out, int out_size, void* d_ws, size_t ws_size,
                              hipStream_t stream)
{
    const float* hidden      = (const float*)d_in[0]; // [64,512,768] f32
    const int*   attn_mask   = (const int*)  d_in[1]; // [64,512] i32
    const int*   labels      = (const int*)  d_in[2]; // [64,512] i32
    const float* W           = (const float*)d_in[3]; // [768,9]
    const float* bias        = (const float*)d_in[4]; // [9]
    const float* start_trans = (const float*)d_in[5]; // [9]
    const float* end_trans   = (const float*)d_in[6]; // [9]
    const float* trans       = (const float*)d_in[7]; // [9,9]

    float* out    = (float*)d_out;   // out[0] = loss, out[1..] = logits flat
    float* logits = out + 1;
    float* llh    = (float*)d_ws;    // 64 floats of scratch

    // 2048 M-tiles of 16 rows, 8 waves (tiles) per 256-thread block.
    bertcrf_logits_wmma<<<256, 256, 0, stream>>>(hidden, W, bias, logits);
    bertcrf_forward<<<BATCH, 32, 0, stream>>>(logits, attn_mask, labels,
                                              start_trans, end_trans, trans, llh);
    bertcrf_loss<<<1, 32, 0, stream>>>(llh, out);
}